// NCVibrationFrontend_59562606461184
// MI455X (gfx1250) — compile-verified
//
#include <hip/hip_runtime.h>
#include <math.h>

// ---------------------------------------------------------------------------
// NCVibrationFrontend on MI455X (gfx1250, wave32, WMMA).
// DFT-as-matmul + filterbank einsums run on V_WMMA_F32_16X16X4_F32.
// ---------------------------------------------------------------------------

typedef __attribute__((ext_vector_type(2))) float v2f;
typedef __attribute__((ext_vector_type(8))) float v8f;

#define B_DIM    32
#define T_SAMP   256000
#define NFFT     512
#define HOP      64
#define NFREQ    257
#define FP       272          // freq padded to 17*16
#define TFRAMES  4001
#define TP       4016         // frames padded to 251*16
#define TTILES   251
#define FBLK     17
#define NBANDS   40
#define NBLK     3            // bands padded to 48
#define KC_DFT   128          // 512 / 4
#define KC_FB    68           // 272 / 4
#define SSTRIDE  514          // LDS frame stride (even, 2-bank skew -> no conflicts)

// workspace offsets in floats
#define OFF_DFTC 0ull
#define SZ_DFT   (17ull*128*64)              // 139264 per matrix
#define OFF_DFTS (OFF_DFTC + SZ_DFT)
#define OFF_FBT  (OFF_DFTS + SZ_DFT)
#define SZ_FBT   (3ull*68*64)                // 13056
#define OFF_MAG  (OFF_FBT + SZ_FBT)
#define SZ_BTF   ((unsigned long long)B_DIM*TP*FP)   // 34,955,264
#define OFF_VNR  (OFF_MAG + SZ_BTF)
#define OFF_BAND (OFF_VNR + SZ_BTF)

static __device__ __forceinline__ float sigm(float x) {
  return 1.0f / (1.0f + expf(-x));
}

// ---------------------------------------------------------------------------
// Init: windowed DFT weights, pre-swizzled into the 16x16x4 B-fragment layout
// (lane l holds B[2*(l/16)+j][l%16] for j=0,1, contiguous float2 per lane).
// ---------------------------------------------------------------------------
__global__ void k_init_dft(const float* __restrict__ window,
                           float* __restrict__ wsC, float* __restrict__ wsS) {
  int tid = blockIdx.x * 256 + threadIdx.x;
  if (tid >= 17 * 128 * 64) return;
  int j     = tid & 1;
  int lane  = (tid >> 1) & 31;
  int chunk = tid >> 6;              // nblk*128 + kc
  int nblk  = chunk / KC_DFT;
  int kc    = chunk % KC_DFT;
  int f = nblk * 16 + (lane & 15);
  int k = kc * 4 + ((lane >> 4) << 1) + j;
  float c = 0.0f, s = 0.0f;
  if (f < NFREQ) {
    float ang = 6.28318530717958647692f * (float)k * (float)f / (float)NFFT;
    float w = window[k];
    c = w * cosf(ang);
    s = w * sinf(ang);
  }
  wsC[tid] = c;
  wsS[tid] = s;
}

// fb^T (272 x 48) pre-swizzled the same way; pad rows/cols are zero.
__global__ void k_init_fbt(const float* __restrict__ fb, float* __restrict__ wsFBT) {
  int tid = blockIdx.x * 256 + threadIdx.x;
  if (tid >= 3 * 68 * 64) return;
  int j     = tid & 1;
  int lane  = (tid >> 1) & 31;
  int chunk = tid >> 6;
  int nblk  = chunk / KC_FB;
  int kc    = chunk % KC_FB;
  int n = nblk * 16 + (lane & 15);
  int f = kc * 4 + ((lane >> 4) << 1) + j;
  wsFBT[tid] = (n < NBANDS && f < NFREQ) ? fb[n * NFREQ + f] : 0.0f;
}

// ---------------------------------------------------------------------------
// STFT magnitude via WMMA DFT matmul. One block = 16 frames, 17 waves each
// owning one 16-freq block; frames staged once in LDS (reflect padding).
// 256 v_wmma_f32_16x16x4_f32 per wave (K=512, real+imag accumulators).
// ---------------------------------------------------------------------------
__global__ __launch_bounds__(544) void k_stft(const float* __restrict__ vib,
                                              const float* __restrict__ wsC,
                                              const float* __restrict__ wsS,
                                              float* __restrict__ mag) {
  __shared__ float sF[16 * SSTRIDE];
  int b     = blockIdx.x / TTILES;
  int ttile = blockIdx.x % TTILES;
  int tid   = threadIdx.x;
  const float* xrow = vib + (size_t)b * T_SAMP;

  for (int i = tid; i < 16 * NFFT; i += 544) {
    int m = i >> 9;
    int k = i & (NFFT - 1);
    int tt = ttile * 16 + m;
    if (tt > TFRAMES - 1) tt = TFRAMES - 1;      // clamp pad rows (discarded later)
    int gi = tt * HOP + k - (NFFT / 2);
    gi = gi < 0 ? -gi : gi;                      // reflect left
    gi = gi >= T_SAMP ? 2 * (T_SAMP - 1) - gi : gi;  // reflect right
    sF[m * SSTRIDE + k] = xrow[gi];
  }
  __syncthreads();

  int wv   = tid >> 5;        // freq block 0..16
  int lane = tid & 31;
  int m    = lane & 15;       // frame row of A fragment
  int koff = (lane >> 4) << 1;
  const float* Wc = wsC + (size_t)wv * KC_DFT * 64;
  const float* Ws = wsS + (size_t)wv * KC_DFT * 64;

  v8f cr = {}; v8f ci = {};
  for (int kc = 0; kc < KC_DFT; ++kc) {
    int k = kc * 4 + koff;
    v2f a  = *(const v2f*)&sF[m * SSTRIDE + k];
    v2f bc = *(const v2f*)(Wc + kc * 64 + lane * 2);
    v2f bs = *(const v2f*)(Ws + kc * 64 + lane * 2);
    cr = __builtin_amdgcn_wmma_f32_16x16x4_f32(false, a, false, bc, (short)0, cr, false, false);
    ci = __builtin_amdgcn_wmma_f32_16x16x4_f32(false, a, false, bs, (short)0, ci, false, false);
  }

  int half = lane >> 4;
  int f = wv * 16 + (lane & 15);
  float* mrow = mag + ((size_t)b * TP + (size_t)ttile * 16) * FP;
#pragma unroll
  for (int v = 0; v < 8; ++v) {
    int mm = v + (half << 3);
    float val = (f < NFREQ) ? sqrtf(cr[v] * cr[v] + ci[v] * ci[v]) : 0.0f;
    mrow[(size_t)mm * FP + f] = val;   // pad freq columns zeroed for einsum K-dim
  }
}

// ---------------------------------------------------------------------------
// Noise-floor tracking scan: one thread per (b,f) chain, coalesced over f.
// Writes vnr[b][t][f] = mag / (|ns|*noise + 1e-8).
// ---------------------------------------------------------------------------
__global__ void k_scan(const float* __restrict__ mag, float* __restrict__ vnr,
                       const float* __restrict__ ns_p, const float* __restrict__ rr_p,
                       const float* __restrict__ rf_p) {
  int tid = blockIdx.x * 256 + threadIdx.x;   // exactly 32*272 threads
  int b = tid / FP;
  int f = tid % FP;
  const float* mrow = mag + (size_t)b * TP * FP + f;
  float* vrow = vnr + (size_t)b * TP * FP + f;
  float rise = sigm(rr_p[0]);
  float fall = sigm(rf_p[0]);
  float ns   = fabsf(ns_p[0]);
  float mn = 3.4e38f;
  for (int t = 0; t < 20; ++t) mn = fminf(mn, mrow[(size_t)t * FP]);
  float nf = fmaxf(mn, 1e-5f);
  float fl = 0.5f * nf;
  for (int t = 0; t < TFRAMES; ++t) {
    float m = mrow[(size_t)t * FP];
    float alpha = (m > nf) ? rise : fall;
    nf = fmaxf(nf + alpha * (m - nf), fl);
    vrow[(size_t)t * FP] = m / (ns * nf + 1e-8f);
  }
}

// ---------------------------------------------------------------------------
// Both filterbank einsums in one pass: band = Mag x FBt, vnr_bands = tanh(0.1 *
// Vnr x FBt). One wave per (b, t-tile, n-block); K=272 -> 68 WMMA pairs.
// ---------------------------------------------------------------------------
__global__ __launch_bounds__(256) void k_band(const float* __restrict__ mag,
                                              const float* __restrict__ vnr,
                                              const float* __restrict__ fbt,
                                              float* __restrict__ band,
                                              float* __restrict__ vnrout) {
  int widx = blockIdx.x * 8 + (threadIdx.x >> 5);   // 0..24095, exact
  int lane = threadIdx.x & 31;
  int b     = widx / (TTILES * NBLK);
  int r     = widx % (TTILES * NBLK);
  int ttile = r / NBLK;
  int nblk  = r % NBLK;
  const float* mA = mag + ((size_t)b * TP + (size_t)ttile * 16) * FP;
  const float* vA = vnr + ((size_t)b * TP + (size_t)ttile * 16) * FP;
  const float* Bw = fbt + (size_t)nblk * KC_FB * 64;
  int m    = lane & 15;
  int koff = (lane >> 4) << 1;

  v8f ab = {}; v8f av = {};
  for (int kc = 0; kc < KC_FB; ++kc) {
    int f0 = kc * 4 + koff;
    v2f am = *(const v2f*)(mA + (size_t)m * FP + f0);
    v2f aw = *(const v2f*)(vA + (size_t)m * FP + f0);
    v2f bb = *(const v2f*)(Bw + kc * 64 + lane * 2);
    ab = __builtin_amdgcn_wmma_f32_16x16x4_f32(false, am, false, bb, (short)0, ab, false, false);
    av = __builtin_amdgcn_wmma_f32_16x16x4_f32(false, aw, false, bb, (short)0, av, false, false);
  }

  int half = lane >> 4;
  int n = nblk * 16 + (lane & 15);
#pragma unroll
  for (int v = 0; v < 8; ++v) {
    int t = ttile * 16 + v + (half << 3);
    if (t < TFRAMES && n < NBANDS) {
      band[((size_t)b * NBANDS + n) * TFRAMES + t] = ab[v];
      float z = tanhf(av[v] * 0.1f);
      if (isnan(z)) z = 0.0f;                    // nan_to_num(nan=0); tanh bounds the rest
      vnrout[((size_t)b * NBANDS + n) * TFRAMES + t] = z;
    }
  }
}

// ---------------------------------------------------------------------------
// Kurtosis gate + standardization: one block per (b,n) row, LDS reductions.
// ---------------------------------------------------------------------------
static __device__ __forceinline__ float blockSum(float v, float* red) {
  int t = threadIdx.x;
  red[t] = v;
  __syncthreads();
#pragma unroll
  for (int s = 128; s >= 1; s >>= 1) {
    if (t < s) red[t] += red[t + s];
    __syncthreads();
  }
  float r = red[0];
  __syncthreads();
  return r;
}

static __device__ __forceinline__ float gatedVal(float bv, float vv, float g) {
  float x = bv * g * (0.5f + 0.5f * vv);
  if (isnan(x)) return 0.0f;
  if (isinf(x)) return x > 0.0f ? 1.0e4f : -1.0e4f;
  return x;
}

__global__ __launch_bounds__(256) void k_gate(const float* __restrict__ band,
                                              const float* __restrict__ vnrout,
                                              float* __restrict__ feat,
                                              const float* __restrict__ gw,
                                              const float* __restrict__ gb,
                                              const float* __restrict__ gf) {
  __shared__ float red[256];
  int rowid = blockIdx.x;                 // b*40 + n
  int n = rowid % NBANDS;
  int tid = threadIdx.x;
  const float* row = band + (size_t)rowid * TFRAMES;
  const float* vr  = vnrout + (size_t)rowid * TFRAMES;
  float* outp = feat + (size_t)rowid * TFRAMES;
  const float invT = 1.0f / (float)TFRAMES;

  float s = 0.0f;
  for (int t = tid; t < TFRAMES; t += 256) s += row[t];
  float mu = blockSum(s, red) * invT;

  float s2 = 0.0f, s4 = 0.0f;
  for (int t = tid; t < TFRAMES; t += 256) {
    float d = row[t] - mu; float d2 = d * d; s2 += d2; s4 += d2 * d2;
  }
  s2 = blockSum(s2, red) * invT;
  s4 = blockSum(s4, red) * invT;
  float var  = fmaxf(s2, 1e-8f);
  float kurt = s4 / (var * var + 1e-8f);
  float kn   = (kurt - 3.0f) * (1.0f / 3.0f);
  float fl   = sigm(gf[n]);
  float g    = sigm(gw[n] * kn + gb[n]);
  g = g * (1.0f - fl) + fl;

  float xs = 0.0f;
  for (int t = tid; t < TFRAMES; t += 256) xs += gatedVal(row[t], vr[t], g);
  float xmu = blockSum(xs, red) * invT;

  float xv = 0.0f;
  for (int t = tid; t < TFRAMES; t += 256) {
    float d = gatedVal(row[t], vr[t], g) - xmu; xv += d * d;
  }
  xv = blockSum(xv, red) * invT;
  float inv = rsqrtf(xv + 1e-5f);

  for (int t = tid; t < TFRAMES; t += 256)
    outp[t] = (gatedVal(row[t], vr[t], g) - xmu) * inv;
}

// ---------------------------------------------------------------------------
extern "C" void kernel_launch(void* const* d_in, const int* in_sizes, int n_in,
                              void* d_out, int out_size, void* d_ws, size_t ws_size,
                              hipStream_t stream) {
  const float* vib = (const float*)d_in[0];
  const float* fb  = (const float*)d_in[1];
  const float* win = (const float*)d_in[2];
  const float* ns  = (const float*)d_in[3];
  const float* rr  = (const float*)d_in[4];
  const float* rf  = (const float*)d_in[5];
  const float* gw  = (const float*)d_in[6];
  const float* gb  = (const float*)d_in[7];
  const float* gf  = (const float*)d_in[8];

  float* ws     = (float*)d_ws;
  float* wsC    = ws + OFF_DFTC;
  float* wsS    = ws + OFF_DFTS;
  float* wsFBT  = ws + OFF_FBT;
  float* wsMag  = ws + OFF_MAG;
  float* wsVnr  = ws + OFF_VNR;
  float* wsBand = ws + OFF_BAND;

  float* feat   = (float*)d_out;                                   // (32,40,4001)
  float* vnrout = feat + (size_t)B_DIM * NBANDS * TFRAMES;         // (32,40,4001)

  k_init_dft<<<(17 * 128 * 64) / 256, 256, 0, stream>>>(win, wsC, wsS);
  k_init_fbt<<<(3 * 68 * 64) / 256, 256, 0, stream>>>(fb, wsFBT);
  k_stft<<<B_DIM * TTILES, 544, 0, stream>>>(vib, wsC, wsS, wsMag);
  k_scan<<<(B_DIM * FP) / 256, 256, 0, stream>>>(wsMag, wsVnr, ns, rr, rf);
  k_band<<<(B_DIM * TTILES * NBLK) / 8, 256, 0, stream>>>(wsMag, wsVnr, wsFBT, wsBand, vnrout);
  k_gate<<<B_DIM * NBANDS, 256, 0, stream>>>(wsBand, vnrout, feat, gw, gb, gf);
}